// GATNet_54855322305119
// MI455X (gfx1250) — compile-verified
//
#include <hip/hip_runtime.h>
#include <math.h>
#include <float.h>

typedef __attribute__((ext_vector_type(2))) float v2f;
typedef __attribute__((ext_vector_type(8))) float v8f;

#define HEADS 4
#define CH 64
#define HC 256          // HEADS*CH
#define NEG_SLOPE 0.2f

// ---------- helpers ----------------------------------------------------------

__device__ __forceinline__ float atomicMaxF(float* addr, float val) {
    // sign-aware int punning: works for any mix of signs, init to -FLT_MAX
    if (val >= 0.0f)
        return __int_as_float(atomicMax((int*)addr, __float_as_int(val)));
    else
        return __uint_as_float(atomicMin((unsigned int*)addr, __float_as_uint(val)));
}

__device__ __forceinline__ void edge_nodes(const int* __restrict__ ei, int e, int E,
                                           int& s, int& d) {
    if (e < E) { s = ei[e]; d = ei[E + e]; }   // edge_index [2,E] row-major
    else       { s = d = e - E; }              // implicit self-loops
}

// ---------- per-layer init ---------------------------------------------------

__global__ void k_init(float* __restrict__ agg, float* __restrict__ emax,
                       float* __restrict__ denom, int N) {
    int i = blockIdx.x * blockDim.x + threadIdx.x;
    if (i < N * HC) agg[i] = 0.0f;
    if (i < N * HEADS) { emax[i] = -FLT_MAX; denom[i] = 0.0f; }
}

// ---------- dense projection: h = in[N,K] @ W[K,256] via V_WMMA_F32_16X16X4_F32
// K is a compile-time constant: K%4==0 -> unconditional vectorized loads and a
// fully unrolled WMMA chain; otherwise zero-pad via selects (no EXEC-predicated
// loads). Store epilogue uses a wave-uniform full-tile fast path.

template <int K>
__global__ void k_proj_wmma(const float* __restrict__ in, const float* __restrict__ W,
                            float* __restrict__ h, int N) {
    int wid  = (blockIdx.x * blockDim.x + threadIdx.x) >> 5;   // global wave id
    int lane = threadIdx.x & 31;
    const int nTiles = HC / 16;                                // 16 col tiles
    int mTile = wid / nTiles;
    int nTile = wid - mTile * nTiles;
    int m0 = mTile * 16;
    if (m0 >= N) return;                                       // wave-uniform exit

    int half = lane >> 4;          // 0: K pair {0,1}; 1: K pair {2,3}
    int l16  = lane & 15;
    int mrow = m0 + l16; if (mrow >= N) mrow = N - 1;          // keep EXEC full
    int ncol = nTile * 16 + l16;

    const float* aBase = in + (size_t)mrow * K;                // A row for this lane
    const float* bBase = W + ncol;                             // B col for this lane

    v8f acc = {};
    #pragma unroll
    for (int k0 = 0; k0 < K; k0 += 4) {
        int ka = k0 + 2 * half;
        float a0, a1, b0, b1;
        if ((K & 3) == 0) {
            // K multiple of 4: unconditional; A pair is contiguous -> b64 load
            float2 av = *(const float2*)(aBase + ka);
            a0 = av.x; a1 = av.y;
            b0 = bBase[(size_t)ka * HC];
            b1 = bBase[(size_t)(ka + 1) * HC];
        } else {
            // zero-pad path: unconditional loads at clamped (valid) addresses,
            // then selects -- no divergent branches
            int i0 = (ka     < K) ? ka     : K - 1;
            int i1 = (ka + 1 < K) ? ka + 1 : K - 1;
            float a0r = aBase[i0], a1r = aBase[i1];
            float b0r = bBase[(size_t)i0 * HC], b1r = bBase[(size_t)i1 * HC];
            a0 = (ka     < K) ? a0r : 0.0f;  b0 = (ka     < K) ? b0r : 0.0f;
            a1 = (ka + 1 < K) ? a1r : 0.0f;  b1 = (ka + 1 < K) ? b1r : 0.0f;
        }
        v2f A; A.x = a0; A.y = a1;
        v2f B; B.x = b0; B.y = b1;
        acc = __builtin_amdgcn_wmma_f32_16x16x4_f32(
            /*neg_a=*/false, A, /*neg_b=*/false, B,
            /*c_mod=*/(short)0, acc, /*reuse_a=*/false, /*reuse_b=*/false);
    }

    // D 16x16 f32: VGPR r -> row m0 + r + 8*half, lane l16 -> col
    float* outBase = h + (size_t)(m0 + 8 * half) * HC + ncol;
    if (m0 + 16 <= N) {
        // full tile (wave-uniform): 8 plain stores with immediate offsets
        #pragma unroll
        for (int r = 0; r < 8; ++r) outBase[(size_t)r * HC] = acc[r];
    } else {
        #pragma unroll
        for (int r = 0; r < 8; ++r) {
            int m = m0 + r + 8 * half;
            if (m < N) h[(size_t)m * HC + ncol] = acc[r];
        }
    }
}

// ---------- attention coefficients per node ---------------------------------

__global__ void k_alpha(const float* __restrict__ h, const float* __restrict__ a_src,
                        const float* __restrict__ a_dst, float* __restrict__ asrc,
                        float* __restrict__ adst, int N) {
    int i = blockIdx.x * blockDim.x + threadIdx.x;     // i = n*HEADS + head
    if (i >= N * HEADS) return;
    int n = i >> 2, hd = i & 3;
    const float* hp = h + (size_t)n * HC + hd * CH;
    const float* s  = a_src + hd * CH;
    const float* d  = a_dst + hd * CH;
    float ss = 0.0f, dd = 0.0f;
    #pragma unroll 8
    for (int c = 0; c < CH; ++c) { float v = hp[c]; ss += v * s[c]; dd += v * d[c]; }
    asrc[i] = ss; adst[i] = dd;
}

// ---------- segment softmax: pass 1 (max), pass 2 (exp + sum) ---------------

__global__ void k_edge_max(const int* __restrict__ ei, const float* __restrict__ asrc,
                           const float* __restrict__ adst, float* __restrict__ emax,
                           int E, int N) {
    int i = blockIdx.x * blockDim.x + threadIdx.x;     // (e,head)
    int total = (E + N) * HEADS;
    if (i >= total) return;
    int e = i >> 2, hd = i & 3;
    int s, d; edge_nodes(ei, e, E, s, d);
    float v = asrc[s * HEADS + hd] + adst[d * HEADS + hd];
    v = v > 0.0f ? v : NEG_SLOPE * v;                  // leaky_relu
    atomicMaxF(&emax[d * HEADS + hd], v);
}

__global__ void k_edge_exp(const int* __restrict__ ei, const float* __restrict__ asrc,
                           const float* __restrict__ adst, const float* __restrict__ emax,
                           float* __restrict__ denom, float* __restrict__ exbuf,
                           int E, int N) {
    int i = blockIdx.x * blockDim.x + threadIdx.x;
    int total = (E + N) * HEADS;
    if (i >= total) return;
    int e = i >> 2, hd = i & 3;
    int s, d; edge_nodes(ei, e, E, s, d);
    float v = asrc[s * HEADS + hd] + adst[d * HEADS + hd];
    v = v > 0.0f ? v : NEG_SLOPE * v;
    float xv = expf(v - emax[d * HEADS + hd]);
    exbuf[i] = xv;
    atomicAdd(&denom[d * HEADS + hd], xv);
}

// ---------- weighted message scatter: one wave32 per (edge, head) -----------

__global__ void k_edge_scatter(const int* __restrict__ ei, const float* __restrict__ h,
                               const float* __restrict__ exbuf, const float* __restrict__ denom,
                               float* __restrict__ agg, int E, int N) {
    int gid  = blockIdx.x * blockDim.x + threadIdx.x;
    int wid  = gid >> 5;
    int lane = gid & 31;
    int total = (E + N) * HEADS;
    if (wid >= total) return;
    int e = wid >> 2, hd = wid & 3;
    int s, d; edge_nodes(ei, e, E, s, d);
    float w = exbuf[wid] / denom[d * HEADS + hd];
    const float* hp = h   + (size_t)s * HC + hd * CH;
    float*       ap = agg + (size_t)d * HC + hd * CH;
    float2 hv = ((const float2*)hp)[lane];             // 64 channels / 32 lanes
    atomicAdd(&ap[2 * lane + 0], hv.x * w);
    atomicAdd(&ap[2 * lane + 1], hv.y * w);
}

// ---------- head-mean + bias + ELU + optional residual -----------------------

__global__ void k_finalize(const float* __restrict__ agg, const float* __restrict__ b,
                           const float* __restrict__ res, float* __restrict__ out, int N) {
    int i = blockIdx.x * blockDim.x + threadIdx.x;     // n*CH + c
    if (i >= N * CH) return;
    int n = i >> 6, c = i & 63;
    const float* ap = agg + (size_t)n * HC;
    float v = 0.25f * (ap[c] + ap[CH + c] + ap[2 * CH + c] + ap[3 * CH + c]) + b[c];
    v = v > 0.0f ? v : expm1f(v);                      // ELU (alpha=1)
    if (res) v += res[i];
    out[i] = v;
}

// ---------- final linear head ------------------------------------------------

__global__ void k_final_linear(const float* __restrict__ h3, const float* __restrict__ lw,
                               const float* __restrict__ lb, float* __restrict__ out, int N) {
    int n = blockIdx.x * blockDim.x + threadIdx.x;
    if (n >= N) return;
    float s = 0.0f;
    #pragma unroll 8
    for (int c = 0; c < CH; ++c) s += h3[(size_t)n * CH + c] * lw[c];
    out[n] = s + lb[0];
}

// ---------- driver -----------------------------------------------------------

extern "C" void kernel_launch(void* const* d_in, const int* in_sizes, int n_in,
                              void* d_out, int out_size, void* d_ws, size_t ws_size,
                              hipStream_t stream) {
    const float* x   = (const float*)d_in[0];
    const int*   ei  = (const int*)  d_in[1];
    const float* W1  = (const float*)d_in[2];
    const float* a1s = (const float*)d_in[3];
    const float* a1d = (const float*)d_in[4];
    const float* b1  = (const float*)d_in[5];
    const float* W2  = (const float*)d_in[6];
    const float* a2s = (const float*)d_in[7];
    const float* a2d = (const float*)d_in[8];
    const float* b2  = (const float*)d_in[9];
    const float* W3  = (const float*)d_in[10];
    const float* a3s = (const float*)d_in[11];
    const float* a3d = (const float*)d_in[12];
    const float* b3  = (const float*)d_in[13];
    const float* lw  = (const float*)d_in[14];
    const float* lb  = (const float*)d_in[15];

    const int N  = in_sizes[0] / 3;     // 50000
    const int E  = in_sizes[1] / 2;     // 800000
    const int EE = E + N;               // + self-loops

    float* p = (float*)d_ws;
    float* h_proj = p; p += (size_t)N * HC;
    float* agg    = p; p += (size_t)N * HC;
    float* asrc   = p; p += (size_t)N * HEADS;
    float* adst   = p; p += (size_t)N * HEADS;
    float* emax   = p; p += (size_t)N * HEADS;
    float* denom  = p; p += (size_t)N * HEADS;
    float* exbuf  = p; p += (size_t)EE * HEADS;
    float* bufA   = p; p += (size_t)N * CH;
    float* bufB   = p; p += (size_t)N * CH;

    const int TB = 256;
    auto blocks = [](long long t, int tb) { return (unsigned)((t + tb - 1) / tb); };

    long long projThreads = (long long)((N + 15) / 16) * (HC / 16) * 32;

    auto layer = [&](const float* in, int Ktag, const float* W, const float* as_,
                     const float* ad_, const float* bb, const float* res, float* out) {
        k_init<<<blocks((long long)N * HC, TB), TB, 0, stream>>>(agg, emax, denom, N);

        if (Ktag == 3)
            k_proj_wmma<3><<<blocks(projThreads, TB), TB, 0, stream>>>(in, W, h_proj, N);
        else
            k_proj_wmma<64><<<blocks(projThreads, TB), TB, 0, stream>>>(in, W, h_proj, N);

        k_alpha<<<blocks((long long)N * HEADS, TB), TB, 0, stream>>>(
            h_proj, as_, ad_, asrc, adst, N);

        k_edge_max<<<blocks((long long)EE * HEADS, TB), TB, 0, stream>>>(
            ei, asrc, adst, emax, E, N);

        k_edge_exp<<<blocks((long long)EE * HEADS, TB), TB, 0, stream>>>(
            ei, asrc, adst, emax, denom, exbuf, E, N);

        k_edge_scatter<<<blocks((long long)EE * HEADS * 32, TB), TB, 0, stream>>>(
            ei, h_proj, exbuf, denom, agg, E, N);

        k_finalize<<<blocks((long long)N * CH, TB), TB, 0, stream>>>(agg, bb, res, out, N);
    };

    layer(x,    3,  W1, a1s, a1d, b1, nullptr, bufA);   // h1 = elu(gat1(x))
    layer(bufA, 64, W2, a2s, a2d, b2, bufA,    bufB);   // h2 = elu(gat2(h1)) + h1
    layer(bufB, 64, W3, a3s, a3d, b3, bufB,    bufA);   // h3 = elu(gat3(h2)) + h2

    k_final_linear<<<blocks(N, TB), TB, 0, stream>>>(bufA, lw, lb, (float*)d_out, N);
}